// Net_66898410602869
// MI455X (gfx1250) — compile-verified
//
#include <hip/hip_runtime.h>

// Problem constants (from reference)
#define T_STEPS 100
#define BATCH   512
#define IN_DIM  784
#define HID_DIM 512
#define OUT_DIM 10
#define BETA    0.95f
#define THRESH  1.0f

#define LSTR 20              // LDS tile stride in floats: 80B (16B aligned, bank-conflict free)

typedef __attribute__((ext_vector_type(2))) float v2f;
typedef __attribute__((ext_vector_type(8))) float v8f;

// ---------------------------------------------------------------------------
// Zero persistent state (mem1, mem2) at the start of every launch.
// ---------------------------------------------------------------------------
__global__ __launch_bounds__(256) void snn_init(float* __restrict__ mem1,
                                                float* __restrict__ mem2) {
    int i = blockIdx.x * 256 + threadIdx.x;
    if (i < BATCH * HID_DIM) mem1[i] = 0.0f;
    if (i < BATCH * OUT_DIM) mem2[i] = 0.0f;
}

// Async 16B global -> LDS copy (ASYNCcnt-tracked; bypasses VGPRs).
__device__ __forceinline__ void async_b128(unsigned lds_byte_addr,
                                           const float* gaddr) {
    asm volatile("global_load_async_to_lds_b128 %0, %1, off"
                 :: "v"(lds_byte_addr), "v"(gaddr) : "memory");
}

// ---------------------------------------------------------------------------
// Layer 1: cur1 = x_t @ w1^T  (B=512 x H=512, K=784), fused Leaky update.
// Block = 4 waves (128 threads), block tile 32x64.
// Wave (mw,nw): 16 rows x 32 cols = two 16x16 WMMA accumulators sharing A.
// K staged in 16-wide chunks, double-buffered LDS filled with
// GLOBAL_LOAD_ASYNC_TO_LDS_B128 (3 async ops/thread/chunk), synced with
// s_wait_asynccnt + workgroup barrier.  f32 WMMA keeps reference precision.
// ---------------------------------------------------------------------------
__global__ __launch_bounds__(128) void snn_layer1(
    const float* __restrict__ x_t,   // [BATCH, IN_DIM] slice at time t
    const float* __restrict__ w1,    // [HID_DIM, IN_DIM]
    const float* __restrict__ b1v,   // [HID_DIM]
    float* __restrict__ mem1,        // [BATCH, HID_DIM] persistent
    float* __restrict__ spk1)        // [BATCH, HID_DIM] spikes (0/1)
{
    __shared__ float lx[2][32 * LSTR];   // X tile: 32 rows x 16 k (padded)
    __shared__ float lw[2][64 * LSTR];   // W tile: 64 h-cols x 16 k (padded)

    const int tid  = threadIdx.x;
    const int wave = tid >> 5;
    const int lane = tid & 31;
    const int half = lane >> 4;          // 0: K pair {0,1}, 1: K pair {2,3}
    const int l    = lane & 15;

    const int bm  = blockIdx.x * 32;     // batch-row base of block
    const int bn  = blockIdx.y * 64;     // hidden-col base of block
    const int wmo = (wave >> 1) << 4;    // wave row offset {0,16}
    const int wno = (wave & 1) << 5;     // wave col offset {0,32}

    // Staging assignment: 16B (4-float) quads.
    const int xr = tid >> 2;             // X row 0..31
    const int wr = tid >> 2;             // W rows wr and wr+32
    const int q4 = (tid & 3) * 4;        // k-quad offset 0,4,8,12

    const unsigned lx_dst[2] = {
        (unsigned)(size_t)&lx[0][xr * LSTR + q4],
        (unsigned)(size_t)&lx[1][xr * LSTR + q4] };
    const unsigned lw_dst0[2] = {
        (unsigned)(size_t)&lw[0][wr * LSTR + q4],
        (unsigned)(size_t)&lw[1][wr * LSTR + q4] };
    const unsigned lw_dst1[2] = {
        (unsigned)(size_t)&lw[0][(wr + 32) * LSTR + q4],
        (unsigned)(size_t)&lw[1][(wr + 32) * LSTR + q4] };

    const float* xg  = x_t + (size_t)(bm + xr) * IN_DIM + q4;
    const float* wg0 = w1  + (size_t)(bn + wr) * IN_DIM + q4;
    const float* wg1 = w1  + (size_t)(bn + wr + 32) * IN_DIM + q4;

    v8f acc0 = {0.0f, 0.0f, 0.0f, 0.0f, 0.0f, 0.0f, 0.0f, 0.0f};
    v8f acc1 = {0.0f, 0.0f, 0.0f, 0.0f, 0.0f, 0.0f, 0.0f, 0.0f};

    const int NCHUNK = IN_DIM / 16;      // 49

    // Prologue: fill buffer 0 with chunk 0 (3 async ops in flight).
    async_b128(lx_dst[0],  xg);
    async_b128(lw_dst0[0], wg0);
    async_b128(lw_dst1[0], wg1);

    for (int c = 0; c < NCHUNK; ++c) {
        const int cur = c & 1;
        const int nxt = cur ^ 1;
        if (c + 1 < NCHUNK) {
            const int kn = (c + 1) * 16;
            async_b128(lx_dst[nxt],  xg  + kn);
            async_b128(lw_dst0[nxt], wg0 + kn);
            async_b128(lw_dst1[nxt], wg1 + kn);
            // 6 in flight; in-order completion => oldest 3 (chunk c) done.
            asm volatile("s_wait_asynccnt 3" ::: "memory");
        } else {
            asm volatile("s_wait_asynccnt 0" ::: "memory");
        }
        __syncthreads();                 // chunk c resident for all waves

        const float* plx = &lx[cur][0];
        const float* plw = &lw[cur][0];
#pragma unroll
        for (int s = 0; s < 4; ++s) {
            const int k0 = s * 4 + 2 * half;
            v2f a, b0, b1;
            a.x  = plx[(wmo + l) * LSTR + k0 + 0];
            a.y  = plx[(wmo + l) * LSTR + k0 + 1];
            b0.x = plw[(wno + l) * LSTR + k0 + 0];
            b0.y = plw[(wno + l) * LSTR + k0 + 1];
            b1.x = plw[(wno + 16 + l) * LSTR + k0 + 0];
            b1.y = plw[(wno + 16 + l) * LSTR + k0 + 1];
            acc0 = __builtin_amdgcn_wmma_f32_16x16x4_f32(
                false, a, false, b0, (short)0, acc0, false, false);
            acc1 = __builtin_amdgcn_wmma_f32_16x16x4_f32(
                false, a, false, b1, (short)0, acc1, false, false);
        }
        __syncthreads();                 // all reads of buf cur done before
    }                                    // it is refilled at iteration c+1

    // acc[r]: lanes 0-15 -> (M=r, N=l), lanes 16-31 -> (M=r+8, N=l)
    const int col0 = bn + wno + l;
    const int col1 = col0 + 16;
    const int mofs = half * 8;
    const float bias0 = b1v[col0];
    const float bias1 = b1v[col1];
#pragma unroll
    for (int r = 0; r < 8; ++r) {
        const int row = bm + wmo + mofs + r;
        {
            const float cur1  = acc0[r] + bias0;
            const float m_old = mem1[row * HID_DIM + col0];
            const float reset = (m_old > THRESH) ? THRESH : 0.0f;
            const float m_new = BETA * m_old + cur1 - reset;
            mem1[row * HID_DIM + col0] = m_new;
            spk1[row * HID_DIM + col0] = ((m_new - THRESH) > 0.0f) ? 1.0f : 0.0f;
        }
        {
            const float cur1  = acc1[r] + bias1;
            const float m_old = mem1[row * HID_DIM + col1];
            const float reset = (m_old > THRESH) ? THRESH : 0.0f;
            const float m_new = BETA * m_old + cur1 - reset;
            mem1[row * HID_DIM + col1] = m_new;
            spk1[row * HID_DIM + col1] = ((m_new - THRESH) > 0.0f) ? 1.0f : 0.0f;
        }
    }
}

// ---------------------------------------------------------------------------
// Layer 2: cur2 = spk1 @ w2^T  (B=512 x O=10, K=512), fused Leaky update.
// One wave per 16-row batch tile; N padded 10 -> 16 with zero w2 columns.
// ---------------------------------------------------------------------------
__global__ __launch_bounds__(32) void snn_layer2(
    const float* __restrict__ spk1,   // [BATCH, HID_DIM]
    const float* __restrict__ w2,     // [OUT_DIM, HID_DIM]
    const float* __restrict__ b2,     // [OUT_DIM]
    float* __restrict__ mem2,         // [BATCH, OUT_DIM] persistent
    float* __restrict__ spk2_out,     // [BATCH, OUT_DIM] at time t
    float* __restrict__ mem2_out)     // [BATCH, OUT_DIM] at time t
{
    const int lane = threadIdx.x & 31;
    const int half = lane >> 4;
    const int l    = lane & 15;
    const int bm   = blockIdx.x * 16;

    v8f acc = {0.0f, 0.0f, 0.0f, 0.0f, 0.0f, 0.0f, 0.0f, 0.0f};

    for (int k0 = 0; k0 < HID_DIM; k0 += 4) {
        const int ka = k0 + 2 * half;
        v2f a, b;
        a.x = spk1[(bm + l) * HID_DIM + ka + 0];
        a.y = spk1[(bm + l) * HID_DIM + ka + 1];
        float bx = 0.0f, by = 0.0f;
        if (l < OUT_DIM) {                 // pad N=10 -> 16 with zeros
            bx = w2[l * HID_DIM + ka + 0];
            by = w2[l * HID_DIM + ka + 1];
        }
        b.x = bx;
        b.y = by;
        acc = __builtin_amdgcn_wmma_f32_16x16x4_f32(
            false, a, false, b, (short)0, acc, false, false);
    }

    if (l < OUT_DIM) {
        const float bias = b2[l];
        const int mofs = half * 8;
#pragma unroll
        for (int r = 0; r < 8; ++r) {
            const int row = bm + mofs + r;
            const float cur   = acc[r] + bias;
            const float m_old = mem2[row * OUT_DIM + l];
            const float reset = (m_old > THRESH) ? THRESH : 0.0f;
            const float m_new = BETA * m_old + cur - reset;
            mem2[row * OUT_DIM + l]     = m_new;
            spk2_out[row * OUT_DIM + l] = ((m_new - THRESH) > 0.0f) ? 1.0f : 0.0f;
            mem2_out[row * OUT_DIM + l] = m_new;
        }
    }
}

// ---------------------------------------------------------------------------
// Host launcher: init state, then 100 x (layer1, layer2) on `stream`.
// Graph-capture safe: no allocation, no sync, fixed launch sequence.
// ---------------------------------------------------------------------------
extern "C" void kernel_launch(void* const* d_in, const int* in_sizes, int n_in,
                              void* d_out, int out_size, void* d_ws, size_t ws_size,
                              hipStream_t stream) {
    const float* x  = (const float*)d_in[0];   // [T, B, I]
    const float* w1 = (const float*)d_in[1];   // [H, I]
    const float* b1 = (const float*)d_in[2];   // [H]
    const float* w2 = (const float*)d_in[3];   // [O, H]
    const float* b2 = (const float*)d_in[4];   // [O]

    float* mem1 = (float*)d_ws;                      // [B, H]
    float* spk1 = mem1 + BATCH * HID_DIM;            // [B, H]
    float* mem2 = spk1 + BATCH * HID_DIM;            // [B, O]

    float* spk2_rec = (float*)d_out;                       // [T, B, O]
    float* mem2_rec = spk2_rec + T_STEPS * BATCH * OUT_DIM;

    snn_init<<<(BATCH * HID_DIM + 255) / 256, 256, 0, stream>>>(mem1, mem2);

    dim3 g1(BATCH / 32, HID_DIM / 64);
    for (int t = 0; t < T_STEPS; ++t) {
        snn_layer1<<<g1, 128, 0, stream>>>(
            x + (size_t)t * BATCH * IN_DIM, w1, b1, mem1, spk1);
        snn_layer2<<<BATCH / 16, 32, 0, stream>>>(
            spk1, w2, b2, mem2,
            spk2_rec + (size_t)t * BATCH * OUT_DIM,
            mem2_rec + (size_t)t * BATCH * OUT_DIM);
    }
}